// GraphSAGE_75204877353213
// MI455X (gfx1250) — compile-verified
//
#include <hip/hip_runtime.h>
#include <cstdint>

#define NN      50000
#define DIN     128
#define DHID    128
#define DOUTF   64
#define BN_EPS  1e-5f

typedef float v2f __attribute__((ext_vector_type(2)));
typedef float v8f __attribute__((ext_vector_type(8)));

// ---------------------------------------------------------------------------
// Dual projection GEMM:  xl = A@Wl.T + bl,  xr = A@Wr.T + br
// One wave computes MT=5 vertically-stacked 16x16 tiles of BOTH outputs with
// V_WMMA_F32_16X16X4_F32.  DK/DOUT are compile-time so every A/B displacement
// (m*16*DK, k) folds into the load's 24-bit immediate offset.  The k-loop is
// software-pipelined with explicit double buffering: the 7 loads of the next
// k-step are issued before the 10 wmma of the current one, so wmma waits on
// loads issued a full step earlier.  Wrap `&(DK-1)` keeps the pipeline
// branch-free (last prefetch harmlessly re-reads k=0).
// xl is additionally written to `agg` (self-loop init for the scatter-mean).
// ---------------------------------------------------------------------------
template <int MT, int DK, int DOUT>
__global__ void __launch_bounds__(256)
sage_gemm_dual(const float* __restrict__ X,
               const float* __restrict__ Wl, const float* __restrict__ bl,
               const float* __restrict__ Wr, const float* __restrict__ br,
               float* __restrict__ xl, float* __restrict__ agg, float* __restrict__ xr,
               int n)
{
    const int wave = threadIdx.x >> 5;
    const int lane = threadIdx.x & 31;
    const int macroCount = n / (16 * MT);          // 50000/80 = 625, exact
    const int macro = blockIdx.x * 8 + wave;
    if (macro >= macroCount) return;               // wave-uniform exit (EXEC all-ones)
    const int tileN = blockIdx.y;

    const int l15   = lane & 15;
    const int khalf = lane >> 4;                   // lanes 16-31 carry K+2,K+3
    const int col   = tileN * 16 + l15;
    const int row0  = macro * (16 * MT) + l15;

    const float* __restrict__ xb = X  + (size_t)row0 * DK + 2 * khalf;
    const float* __restrict__ wl = Wl + (size_t)col  * DK + 2 * khalf;
    const float* __restrict__ wr = Wr + (size_t)col  * DK + 2 * khalf;

    v8f accL[MT];
    v8f accR[MT];
#pragma unroll
    for (int m = 0; m < MT; ++m) { accL[m] = (v8f){}; accR[m] = (v8f){}; }

    // pipeline prologue: fragments for k = 0
    v2f bL0 = *(const v2f*)(wl);
    v2f bR0 = *(const v2f*)(wr);
    v2f a0[MT];
#pragma unroll
    for (int m = 0; m < MT; ++m) a0[m] = *(const v2f*)(xb + m * 16 * DK);
    v2f bL1, bR1, a1[MT];

#pragma unroll 2
    for (int k = 0; k < DK; k += 8) {
        // prefetch k+4 fragments (always in range: k+4 <= DK-4)
        bL1 = *(const v2f*)(wl + k + 4);
        bR1 = *(const v2f*)(wr + k + 4);
#pragma unroll
        for (int m = 0; m < MT; ++m)
            a1[m] = *(const v2f*)(xb + m * 16 * DK + k + 4);

        // compute step k from buffer 0
#pragma unroll
        for (int m = 0; m < MT; ++m) {
            accL[m] = __builtin_amdgcn_wmma_f32_16x16x4_f32(false, a0[m], false, bL0,
                                                            (short)0, accL[m], false, false);
            accR[m] = __builtin_amdgcn_wmma_f32_16x16x4_f32(false, a0[m], false, bR0,
                                                            (short)0, accR[m], false, false);
        }

        // prefetch k+8 fragments (wraps to 0 on the final step: harmless reload)
        const int kn = (k + 8) & (DK - 1);
        bL0 = *(const v2f*)(wl + kn);
        bR0 = *(const v2f*)(wr + kn);
#pragma unroll
        for (int m = 0; m < MT; ++m)
            a0[m] = *(const v2f*)(xb + m * 16 * DK + kn);

        // compute step k+4 from buffer 1
#pragma unroll
        for (int m = 0; m < MT; ++m) {
            accL[m] = __builtin_amdgcn_wmma_f32_16x16x4_f32(false, a1[m], false, bL1,
                                                            (short)0, accL[m], false, false);
            accR[m] = __builtin_amdgcn_wmma_f32_16x16x4_f32(false, a1[m], false, bR1,
                                                            (short)0, accR[m], false, false);
        }
    }

    const float biasL = bl[col];
    const float biasR = br[col];
#pragma unroll
    for (int m = 0; m < MT; ++m) {
#pragma unroll
        for (int v = 0; v < 8; ++v) {
            const int    r = macro * (16 * MT) + m * 16 + v + 8 * khalf;
            const size_t o = (size_t)r * DOUT + col;
            const float cl = accL[m][v] + biasL;
            xl[o]  = cl;
            agg[o] = cl;                           // self-loop contribution
            xr[o]  = accR[m][v] + biasR;
        }
    }
}

// ---------------------------------------------------------------------------
// BN fold into the next layer's weights:  (x*sc+sh)@W.T = x@(W*sc).T + W@sh
// One block per output row j: Wout[j,k] = W[j,k]*scale[k],
// bout[j] = bias[j] + sum_k W[j,k]*shift[k].   blockDim = 128 (= din).
// ---------------------------------------------------------------------------
__global__ void __launch_bounds__(128)
fold_bn_weights(const float* __restrict__ W, const float* __restrict__ bias,
                const float* __restrict__ scale, const float* __restrict__ shift,
                float* __restrict__ Wout, float* __restrict__ bout, int din)
{
    const int j = blockIdx.x;
    const int k = threadIdx.x;
    const float w = W[(size_t)j * din + k];
    Wout[(size_t)j * din + k] = w * scale[k];
    float dotp = w * shift[k];
#pragma unroll
    for (int off = 16; off >= 1; off >>= 1)
        dotp += __shfl_xor(dotp, off, 32);
    __shared__ float red[4];
    if ((k & 31) == 0) red[k >> 5] = dotp;
    __syncthreads();
    if (k == 0) bout[j] = bias[j] + red[0] + red[1] + red[2] + red[3];
}

// ---------------------------------------------------------------------------
// Edge scatter:  agg[dst] += xl[src]   (float4 chunks, fp32 HW atomics in L2)
// ---------------------------------------------------------------------------
__global__ void __launch_bounds__(256)
sage_scatter(const float* __restrict__ xl, const int* __restrict__ src,
             const int* __restrict__ dst, float* __restrict__ agg,
             long long nthreads, int d, int chunk_shift)
{
    const long long gid = (long long)blockIdx.x * blockDim.x + threadIdx.x;
    if (gid >= nthreads) return;
    const long long e  = gid >> chunk_shift;
    const int       c4 = (int)(gid & ((1LL << chunk_shift) - 1)) << 2;
    const int s = src[e];
    const int t = dst[e];
    const float4 v = *(const float4*)(xl + (size_t)s * d + c4);
    float* p = agg + (size_t)t * d + c4;
    unsafeAtomicAdd(p + 0, v.x);
    unsafeAtomicAdd(p + 1, v.y);
    unsafeAtomicAdd(p + 2, v.z);
    unsafeAtomicAdd(p + 3, v.w);
}

__global__ void init_cnt(float* __restrict__ cnt, int n)
{
    int i = blockIdx.x * blockDim.x + threadIdx.x;
    if (i < n) cnt[i] = 1.0f;                      // self loop
}

__global__ void count_edges(const int* __restrict__ dst, float* __restrict__ cnt, int E)
{
    int e = blockIdx.x * blockDim.x + threadIdx.x;
    if (e < E) unsafeAtomicAdd(&cnt[dst[e]], 1.0f);
}

__global__ void zero_stats(float* __restrict__ stats, int n)
{
    int i = blockIdx.x * blockDim.x + threadIdx.x;
    if (i < n) stats[i] = 0.0f;
}

// ---------------------------------------------------------------------------
// Combine (h = agg/cnt + xr) fused with per-column sum / sum-of-squares
// accumulation for batchnorm.  blockDim.x == d, one block covers 256 rows.
// ---------------------------------------------------------------------------
__global__ void
combine_stats(const float* __restrict__ agg, const float* __restrict__ xr,
              const float* __restrict__ cnt, float* __restrict__ h,
              float* __restrict__ stats, int n, int d, int do_stats)
{
    const int col = threadIdx.x;
    const int r0  = blockIdx.x * 256;
    int r1 = r0 + 256; if (r1 > n) r1 = n;
    float s = 0.0f, s2 = 0.0f;
    for (int r = r0; r < r1; ++r) {
        const size_t o = (size_t)r * d + col;
        const float v = agg[o] / cnt[r] + xr[o];
        h[o] = v;
        s  += v;
        s2 += v * v;
    }
    if (do_stats) {
        unsafeAtomicAdd(&stats[col],     s);
        unsafeAtomicAdd(&stats[d + col], s2);
    }
}

__global__ void
finalize_bn(const float* __restrict__ stats,
            const float* __restrict__ gamma, const float* __restrict__ beta,
            float* __restrict__ scale, float* __restrict__ shift, int n, int d)
{
    const int c = threadIdx.x;
    if (c >= d) return;
    const float inv_n = 1.0f / (float)n;
    const float mean  = stats[c] * inv_n;
    const float var   = stats[d + c] * inv_n - mean * mean;  // biased variance
    const float sc    = gamma[c] * rsqrtf(var + BN_EPS);
    scale[c] = sc;
    shift[c] = beta[c] - mean * sc;
}

// ---------------------------------------------------------------------------
// Row-wise log_softmax over 64 columns; one wave32 per row (2 cols per lane).
// Writes log_softmax(h) and h (tuple output, concatenated flat).
// ---------------------------------------------------------------------------
__global__ void __launch_bounds__(256)
logsoftmax64(const float* __restrict__ h, float* __restrict__ out_ls,
             float* __restrict__ out_h, int n)
{
    const int wave = threadIdx.x >> 5;
    const int lane = threadIdx.x & 31;
    const int row  = blockIdx.x * 8 + wave;
    if (row >= n) return;
    const size_t o = (size_t)row * 64;
    const float a = h[o + lane];
    const float b = h[o + lane + 32];
    float m = fmaxf(a, b);
#pragma unroll
    for (int off = 16; off >= 1; off >>= 1)
        m = fmaxf(m, __shfl_xor(m, off, 32));
    float s = __expf(a - m) + __expf(b - m);
#pragma unroll
    for (int off = 16; off >= 1; off >>= 1)
        s += __shfl_xor(s, off, 32);
    const float lse = m + __logf(s);
    out_ls[o + lane]      = a - lse;
    out_ls[o + lane + 32] = b - lse;
    out_h[o + lane]       = a;
    out_h[o + lane + 32]  = b;
}

// ---------------------------------------------------------------------------
extern "C" void kernel_launch(void* const* d_in, const int* in_sizes, int n_in,
                              void* d_out, int out_size, void* d_ws, size_t ws_size,
                              hipStream_t stream)
{
    const float* x   = (const float*)d_in[0];
    const int*   ei  = (const int*)  d_in[1];
    const float* Wl0 = (const float*)d_in[2];
    const float* bl0 = (const float*)d_in[3];
    const float* Wr0 = (const float*)d_in[4];
    const float* br0 = (const float*)d_in[5];
    const float* Wl1 = (const float*)d_in[6];
    const float* bl1 = (const float*)d_in[7];
    const float* Wr1 = (const float*)d_in[8];
    const float* br1 = (const float*)d_in[9];
    const float* Wl2 = (const float*)d_in[10];
    const float* bl2 = (const float*)d_in[11];
    const float* Wr2 = (const float*)d_in[12];
    const float* br2 = (const float*)d_in[13];
    const float* g0  = (const float*)d_in[14];
    const float* be0 = (const float*)d_in[15];
    const float* g1  = (const float*)d_in[16];
    const float* be1 = (const float*)d_in[17];

    const int n = NN;
    const int E = in_sizes[1] / 2;
    const int* src = ei;
    const int* dst = ei + E;

    float* ws = (float*)d_ws;
    const size_t ND = (size_t)n * DHID;
    float* xl    = ws;                       // [n,128]
    float* xr    = ws + ND;                  // [n,128]
    float* agg   = ws + 2 * ND;              // [n,128]
    float* h     = ws + 3 * ND;              // [n,128]
    float* cnt   = ws + 4 * ND;              // [n]
    float* stats = cnt + n;                  // [2*128]
    float* scale = stats + 2 * DHID;         // [128]
    float* shift = scale + DHID;             // [128]
    float* Wls   = shift + DHID;             // [128*128] BN-folded left weights
    float* Wrs   = Wls + DHID * DHID;        // [128*128] BN-folded right weights
    float* bls   = Wrs + DHID * DHID;        // [128]
    float* brs   = bls + DHID;               // [128]

    // degree counts (indegree + self loop): identical for all 3 layers
    init_cnt<<<(n + 255) / 256, 256, 0, stream>>>(cnt, n);
    count_edges<<<(E + 255) / 256, 256, 0, stream>>>(dst, cnt, E);

    constexpr int MT = 5;
    const int macroCount = n / (16 * MT);              // 625
    dim3 gridHid((macroCount + 7) / 8, DHID / 16);
    dim3 gridOut((macroCount + 7) / 8, DOUTF / 16);
    const long long st128 = (long long)E * (DHID / 4);
    const long long st64  = (long long)E * (DOUTF / 4);

    // ---- layer 0: SAGEConv(128->128) + BN ----
    sage_gemm_dual<MT, DIN, DHID><<<gridHid, 256, 0, stream>>>(
        x, Wl0, bl0, Wr0, br0, xl, agg, xr, n);
    sage_scatter<<<(int)((st128 + 255) / 256), 256, 0, stream>>>(xl, src, dst, agg, st128, DHID, 5);
    zero_stats<<<1, 256, 0, stream>>>(stats, 2 * DHID);
    combine_stats<<<(n + 255) / 256, DHID, 0, stream>>>(agg, xr, cnt, h, stats, n, DHID, 1);
    finalize_bn<<<1, DHID, 0, stream>>>(stats, g0, be0, scale, shift, n, DHID);

    // ---- layer 1: SAGEConv(128->128) + BN (BN0 folded into weights) ----
    fold_bn_weights<<<DHID, DHID, 0, stream>>>(Wl1, bl1, scale, shift, Wls, bls, DHID);
    fold_bn_weights<<<DHID, DHID, 0, stream>>>(Wr1, br1, scale, shift, Wrs, brs, DHID);
    sage_gemm_dual<MT, DHID, DHID><<<gridHid, 256, 0, stream>>>(
        h, Wls, bls, Wrs, brs, xl, agg, xr, n);
    sage_scatter<<<(int)((st128 + 255) / 256), 256, 0, stream>>>(xl, src, dst, agg, st128, DHID, 5);
    zero_stats<<<1, 256, 0, stream>>>(stats, 2 * DHID);
    combine_stats<<<(n + 255) / 256, DHID, 0, stream>>>(agg, xr, cnt, h, stats, n, DHID, 1);
    finalize_bn<<<1, DHID, 0, stream>>>(stats, g1, be1, scale, shift, n, DHID);

    // ---- layer 2: SAGEConv(128->64) (BN1 folded into weights) ----
    fold_bn_weights<<<DOUTF, DHID, 0, stream>>>(Wl2, bl2, scale, shift, Wls, bls, DHID);
    fold_bn_weights<<<DOUTF, DHID, 0, stream>>>(Wr2, br2, scale, shift, Wrs, brs, DHID);
    sage_gemm_dual<MT, DHID, DOUTF><<<gridOut, 256, 0, stream>>>(
        h, Wls, bls, Wrs, brs, xl, agg, xr, n);
    sage_scatter<<<(int)((st64 + 255) / 256), 256, 0, stream>>>(xl, src, dst, agg, st64, DOUTF, 4);
    combine_stats<<<(n + 255) / 256, DOUTF, 0, stream>>>(agg, xr, cnt, h, stats, n, DOUTF, 0);

    // ---- outputs: (log_softmax(h), h) concatenated ----
    float* out = (float*)d_out;
    logsoftmax64<<<(n + 7) / 8, 256, 0, stream>>>(h, out, out + (size_t)n * DOUTF, n);
}